// SpectralPatchEmbed_56676388438044
// MI455X (gfx1250) — compile-verified
//
#include <hip/hip_runtime.h>
#include <hip/hip_bf16.h>

typedef float v2f __attribute__((ext_vector_type(2)));
typedef float v8f __attribute__((ext_vector_type(8)));

#define BATCH 128
#define CHAN  3
#define HW    224
#define PSZ   16
#define PWID  9
#define NPAT  196      // (224/16)^2
#define EMB   768
#define KDIM  768      // CHAN * 16 * 16
#define FDIM  432      // CHAN * 16 * 9
#define MROWS 25088    // BATCH * NPAT

// ---------------------------------------------------------------------------
// Kernel 1: build effective weight weff[e][k]  (natural layout: k contiguous,
// so GEMM B-fragments {k, k+1} are adjacent -> single ds_load_b64).
//   weff[e][k=(c,n1,n2)] = sum_{k1,k2} proj_w[e, c*144+k1*9+k2]
//                          * fwh[k1]*fww[k2]*cos(2pi(k1*n1+k2*n2)/16)/16
// One block per e; proj_w row staged in LDS (broadcast reads), stores coalesced.
// ---------------------------------------------------------------------------
__global__ __launch_bounds__(256) void build_weff_kernel(
    const float* __restrict__ proj_w, const float* __restrict__ fwh,
    const float* __restrict__ fww, float* __restrict__ weff)
{
  __shared__ float wrow[FDIM];
  __shared__ float ctab[16];
  const int tid = threadIdx.x;
  const int e = blockIdx.x;

  if (tid < FDIM)        wrow[tid]       = proj_w[(size_t)e * FDIM + tid];
  if (tid + 256 < FDIM)  wrow[tid + 256] = proj_w[(size_t)e * FDIM + tid + 256];
  if (tid < 16) ctab[tid] = __cosf((float)tid * 0.39269908169872414f) * 0.0625f; // 2pi/16, ortho 1/16
  __syncthreads();
  // fold frequency weights into the staged row
  if (tid < FDIM) {
    int fl = tid % 144; wrow[tid] *= fwh[fl / 9] * fww[fl % 9];
  }
  if (tid + 256 < FDIM) {
    int f = tid + 256; int fl = f % 144; wrow[f] *= fwh[fl / 9] * fww[fl % 9];
  }
  __syncthreads();

  for (int kk = tid; kk < KDIM; kk += 256) {
    const int c   = kk >> 8;
    const int pix = kk & 255;
    const int n1  = pix >> 4;
    const int n2  = pix & 15;
    const float* wr = &wrow[c * 144];
    float acc = 0.f;
    #pragma unroll
    for (int k1 = 0; k1 < 16; ++k1) {
      const int mb = (k1 * n1) & 15;
      #pragma unroll
      for (int k2 = 0; k2 < 9; ++k2) {
        acc += wr[k1 * 9 + k2] * ctab[(mb + k2 * n2) & 15];
      }
    }
    weff[(size_t)e * KDIM + kk] = acc;   // coalesced in kk
  }
}

// ---------------------------------------------------------------------------
// Kernel 2: GEMM  out[m, e] = sum_k A[m,k] * weff[e,k]  + proj_b[e] + pos_emb
//   m = b*196 + n (patch), k = c*256 + p1*16 + p2 (patch pixel)
// Block: 256 threads = 8 wave32.  Block tile 128(M) x 128(E), K chunks of 32.
// Wave grid 4(M) x 2(E): each wave owns 32M (2 strips) x 64E (4 tiles)
// = 8 v8f accumulators; B fragments reused across both M strips.
// Fragments are single aligned ds_load_b64, software-pipelined 1 K-step ahead.
// ---------------------------------------------------------------------------
__global__ __launch_bounds__(256) void spectral_gemm_kernel(
    const float* __restrict__ x, const float* __restrict__ weff,
    const float* __restrict__ proj_b, const float* __restrict__ pos_emb,
    float* __restrict__ out)
{
  __shared__ float As[128][36];   // 128 patch-rows x 32-k chunk (+pad)
  __shared__ float Bs[128][36];   // 128 e-rows    x 32-k chunk (+pad)

  const int tid   = threadIdx.x;
  const int lane  = tid & 31;
  const int wave  = tid >> 5;
  const int wm    = wave & 3;           // M quarter (32 rows)
  const int we    = wave >> 2;          // E half (64 cols)
  const int lrow  = lane & 15;
  const int khalf = (lane >> 4) << 1;   // 0 or 2 (K split across half-waves)

  const int m_tile = blockIdx.x * 128;
  const int e_tile = blockIdx.y * 128;

  v8f acc[2][4] = {};

  for (int kc0 = 0; kc0 < KDIM; kc0 += 32) {
    // ---- stage A: 128 patches x 32 pixels (2 patch rows) of this K chunk --
    {
      const int c      = kc0 >> 8;            // channel (chunk never straddles)
      const int p1base = (kc0 & 255) >> 4;    // first patch row of chunk
      #pragma unroll
      for (int i = 0; i < 4; ++i) {
        const int idx     = tid + i * 256;    // 0..1023 float4 slots
        const int m_local = idx >> 3;
        const int sub     = idx & 7;
        const int p1l     = sub >> 2;
        const int quad    = sub & 3;
        const int m  = m_tile + m_local;
        const int b  = m / NPAT;
        const int n  = m - b * NPAT;
        const int ph = n / 14;
        const int pw = n - ph * 14;
        const size_t gaddr =
            ((size_t)(b * CHAN + c) * HW + (ph * PSZ + p1base + p1l)) * HW
            + pw * PSZ + quad * 4;
        const float4 v = *(const float4*)&x[gaddr];
        *(float4*)&As[m_local][p1l * 16 + quad * 4] = v;
      }
    }
    // ---- stage B: 128 e-rows x 32 k of weff (row-contiguous, coalesced) ---
    #pragma unroll
    for (int i = 0; i < 4; ++i) {
      const int idx  = tid + i * 256;         // 0..1023 float4 slots
      const int erow = idx >> 3;
      const int col4 = (idx & 7) << 2;
      const float4 v = *(const float4*)&weff[(size_t)(e_tile + erow) * KDIM + kc0 + col4];
      *(float4*)&Bs[erow][col4] = v;
    }
    __syncthreads();

    // ---- 8 K-steps of 4; 8 WMMAs per step; frags pipelined 1 step ahead ---
    v2f a_c[2], b_c[4];
    #pragma unroll
    for (int s = 0; s < 2; ++s)
      a_c[s] = *(const v2f*)&As[wm * 32 + s * 16 + lrow][khalf];
    #pragma unroll
    for (int t = 0; t < 4; ++t)
      b_c[t] = *(const v2f*)&Bs[we * 64 + t * 16 + lrow][khalf];

    #pragma unroll
    for (int kk = 0; kk < 32; kk += 4) {
      v2f a_n[2], b_n[4];
      if (kk < 28) {
        #pragma unroll
        for (int s = 0; s < 2; ++s)
          a_n[s] = *(const v2f*)&As[wm * 32 + s * 16 + lrow][kk + 4 + khalf];
        #pragma unroll
        for (int t = 0; t < 4; ++t)
          b_n[t] = *(const v2f*)&Bs[we * 64 + t * 16 + lrow][kk + 4 + khalf];
      }
      #pragma unroll
      for (int s = 0; s < 2; ++s) {
        #pragma unroll
        for (int t = 0; t < 4; ++t) {
          acc[s][t] = __builtin_amdgcn_wmma_f32_16x16x4_f32(
              false, a_c[s], false, b_c[t], (short)0, acc[s][t], false, false);
        }
      }
      if (kk < 28) {
        #pragma unroll
        for (int s = 0; s < 2; ++s) a_c[s] = a_n[s];
        #pragma unroll
        for (int t = 0; t < 4; ++t) b_c[t] = b_n[t];
      }
    }
    __syncthreads();
  }

  // ---- epilogue: + bias + pos_emb, scatter into (B, 197, 768) -------------
  #pragma unroll
  for (int t = 0; t < 4; ++t) {
    const int e    = e_tile + we * 64 + t * 16 + lrow;
    const float bv = proj_b[e];
    #pragma unroll
    for (int s = 0; s < 2; ++s) {
      #pragma unroll
      for (int r = 0; r < 8; ++r) {
        const int m_local = wm * 32 + s * 16 + r + ((lane >> 4) << 3);
        const int m = m_tile + m_local;
        const int b = m / NPAT;
        const int n = m - b * NPAT;
        const float val = acc[s][t][r] + bv + pos_emb[(size_t)(1 + n) * EMB + e];
        out[((size_t)b * (NPAT + 1) + 1 + n) * EMB + e] = val;
      }
    }
  }
}

// ---------------------------------------------------------------------------
// Kernel 3: cls row  out[b, 0, e] = cls_token[e] + pos_emb[0, e]
// ---------------------------------------------------------------------------
__global__ __launch_bounds__(256) void cls_row_kernel(
    const float* __restrict__ cls_token, const float* __restrict__ pos_emb,
    float* __restrict__ out)
{
  const int t = blockIdx.x * 256 + threadIdx.x;   // 0 .. 128*768-1
  const int b = t / EMB;
  const int e = t - b * EMB;
  out[(size_t)b * (NPAT + 1) * EMB + e] = cls_token[e] + pos_emb[e];
}

extern "C" void kernel_launch(void* const* d_in, const int* in_sizes, int n_in,
                              void* d_out, int out_size, void* d_ws, size_t ws_size,
                              hipStream_t stream) {
  const float* x      = (const float*)d_in[0];
  const float* fwh    = (const float*)d_in[1];
  const float* fww    = (const float*)d_in[2];
  const float* proj_w = (const float*)d_in[3];
  const float* proj_b = (const float*)d_in[4];
  const float* cls_t  = (const float*)d_in[5];
  const float* pos    = (const float*)d_in[6];
  float* out  = (float*)d_out;
  float* weff = (float*)d_ws;   // 768*768 f32 = 2.25 MB effective weight

  build_weff_kernel<<<EMB, 256, 0, stream>>>(proj_w, fwh, fww, weff);
  spectral_gemm_kernel<<<dim3(MROWS / 128, EMB / 128), 256, 0, stream>>>(
      x, weff, proj_b, pos, out);
  cls_row_kernel<<<(BATCH * EMB) / 256, 256, 0, stream>>>(cls_t, pos, out);
}